// WarpingLayer_3stacks_71227737637108
// MI455X (gfx1250) — compile-verified
//
#include <hip/hip_runtime.h>

// Problem geometry (fixed by the reference).
#define DIN   384
#define HIN   128
#define WIN   128
#define NPTS  (DIN * HIN * WIN)     // 6,291,456 input points
#define S     128                   // output volume is 128^3
#define S2    (S * S)
#define S3    (S * S * S)           // per-channel output stride

// Chunking: each block owns 8192 contiguous points = 64 x-rows at fixed z.
#define CHUNK  8192
#define ROWS   (CHUNK / WIN)         // 64 rows of y per block
#define PAD    4                     // covers |flow_y| <= 4 (~99.994% of N(0,1))
#define TROWS  (ROWS + 2 * PAD + 1)  // 73 tile rows
#define TWORDS (TROWS * WIN)         // 9344 words per channel (tile = 74.75 KB total)

// Non-returning f32 atomic add, device scope, relaxed -> global_atomic_add_f32.
__device__ __forceinline__ void atomic_add_f32(float* p, float v) {
    (void)__hip_atomic_fetch_add(p, v, __ATOMIC_RELAXED, __HIP_MEMORY_SCOPE_AGENT);
}

// LDS f32 atomic add, workgroup scope -> ds_add_f32 (addrspace inferred from __shared__).
__device__ __forceinline__ void lds_add_f32(float* p, float v) {
    (void)__hip_atomic_fetch_add(p, v, __ATOMIC_RELAXED, __HIP_MEMORY_SCOPE_WORKGROUP);
}

__global__ __launch_bounds__(256) void zero_out_kernel(float4* __restrict__ out, int n4) {
    int i = blockIdx.x * 256 + threadIdx.x;
    if (i < n4) out[i] = float4{0.0f, 0.0f, 0.0f, 0.0f};
}

__global__ __launch_bounds__(256) void splat_kernel(
    const float* __restrict__ xin,
    const float* __restrict__ flow,
    const float* __restrict__ maskin,
    float* __restrict__ out)
{
    // Private accumulator band of the contended z=127 output plane.
    // Layout: [ch0: TROWS x 128][ch1: TROWS x 128]
    __shared__ float smem[2 * TWORDS];

    const int t    = threadIdx.x;
    const int base = blockIdx.x * CHUNK;       // contiguous chunk, fixed input z
    const int z    = base >> 14;               // / (128*128)
    const int y0   = (base & 16383) >> 7;      // 0 or 64
    const int ty0  = max(y0 - PAD, 0);         // first plane row covered by the tile

    // Zero the LDS tile.
    for (int w = t; w < 2 * TWORDS; w += 256) smem[w] = 0.0f;
    __syncthreads();

    for (int k = 0; k < CHUNK / 256; ++k) {
        const int i = base + k * 256 + t;

        // Streaming prefetch (gfx1250 global_prefetch_b8): one lane per wave32,
        // 8 iterations (2048 elements) ahead per stream.
        if ((t & 31) == 0) {
            const int pf = i + 2048;
            __builtin_prefetch(flow + pf,            0, 0);
            __builtin_prefetch(flow + NPTS + pf,     0, 0);
            __builtin_prefetch(flow + 2 * NPTS + pf, 0, 0);
            __builtin_prefetch(xin + pf,             0, 0);
            __builtin_prefetch(maskin + pf,          0, 0);
        }

        const int y   = (i & 16383) >> 7;
        const int col = i & 127;

        // Flow rescale is identity ((d-1)/(d-1) == 1 for all dims).
        const float m  = maskin[i];
        const float v  = xin[i] * m;            // ch0 = x*mask, ch1 = mask
        const float gz = flow[i]            + (float)z;
        const float gy = flow[i + NPTS]     + (float)y;
        const float gx = flow[i + 2 * NPTS] + (float)col;

        const float fz = floorf(gz); const float tz = gz - fz; const int iz = (int)fz;
        const float fy = floorf(gy); const float ty = gy - fy; const int iy = (int)fy;
        const float fx = floorf(gx); const float tx = gx - fx; const int ix = (int)fx;

        const int izA = min(max(iz,     0), S - 1);
        const int izB = min(max(iz + 1, 0), S - 1);
        const int iyA = min(max(iy,     0), S - 1);
        const int iyB = min(max(iy + 1, 0), S - 1);
        const int ixA = min(max(ix,     0), S - 1);
        const int ixB = min(max(ix + 1, 0), S - 1);

        float wz0 = 1.0f - tz, wz1 = tz;
        float wy0 = 1.0f - ty, wy1 = ty;
        float wx0 = 1.0f - tx, wx1 = tx;

        // Dedup: coincident clamped corners carry total weight exactly 1.
        if (izA == izB) { wz0 = 1.0f; wz1 = 0.0f; }
        if (iyA == iyB) { wy0 = 1.0f; wy1 = 0.0f; }
        if (ixA == ixB) { wx0 = 1.0f; wx1 = 0.0f; }

        const float wzyAA = wz0 * wy0;
        const float wzyAB = wz0 * wy1;
        const float wzyBA = wz1 * wy0;
        const float wzyBB = wz1 * wy1;

        // Exact partition: z=127 corners inside the tile band -> LDS; else global.
        auto emit = [&](int izc, int iyc, int ixc, float w) {
            if (w == 0.0f) return;
            const float a = v * w;
            const float b = m * w;
            if (izc == S - 1) {
                const unsigned ry = (unsigned)(iyc - ty0);
                if (ry < (unsigned)TROWS) {
                    const int l = (int)ry * WIN + ixc;
                    lds_add_f32(&smem[l], a);
                    lds_add_f32(&smem[TWORDS + l], b);
                    return;
                }
            }
            const int idx = izc * S2 + iyc * S + ixc;
            atomic_add_f32(out + idx,      a);
            atomic_add_f32(out + S3 + idx, b);
        };

        emit(izA, iyA, ixA, wzyAA * wx0);
        emit(izA, iyA, ixB, wzyAA * wx1);
        emit(izA, iyB, ixA, wzyAB * wx0);
        emit(izA, iyB, ixB, wzyAB * wx1);
        emit(izB, iyA, ixA, wzyBA * wx0);
        emit(izB, iyA, ixB, wzyBA * wx1);
        emit(izB, iyB, ixA, wzyBB * wx0);
        emit(izB, iyB, ixB, wzyBB * wx1);
    }

    __syncthreads();

    // Flush nonzero tile words into the global z=127 plane.
    const int planeBase = (S - 1) * S2;
    for (int w = t; w < TWORDS; w += 256) {
        const int ry  = w >> 7;
        const int col = w & 127;
        const int gy  = ty0 + ry;
        if (gy <= S - 1) {
            const float a = smem[w];
            const float b = smem[TWORDS + w];
            const int idx = planeBase + gy * S + col;
            if (a != 0.0f) atomic_add_f32(out + idx,      a);
            if (b != 0.0f) atomic_add_f32(out + S3 + idx, b);
        }
    }
}

extern "C" void kernel_launch(void* const* d_in, const int* in_sizes, int n_in,
                              void* d_out, int out_size, void* d_ws, size_t ws_size,
                              hipStream_t stream) {
    (void)in_sizes; (void)n_in; (void)d_ws; (void)ws_size;

    const float* x    = (const float*)d_in[0];   // [1,1,384,128,128]
    const float* flow = (const float*)d_in[1];   // [1,3,384,128,128]
    const float* mask = (const float*)d_in[2];   // [1,1,384,128,128]
    float*       out  = (float*)d_out;           // [1,2,128,128,128]

    // 1) Zero the output accumulator (harness poisons it with 0xAA).
    const int n4 = out_size / 4;                 // 4,194,304 floats -> 1,048,576 float4
    zero_out_kernel<<<(n4 + 255) / 256, 256, 0, stream>>>((float4*)out, n4);

    // 2) Scatter-splat with LDS privatization of the contended z=127 plane.
    splat_kernel<<<NPTS / CHUNK, 256, 0, stream>>>(x, flow, mask, out);
}